// Autoencoder_19696720019683
// MI455X (gfx1250) — compile-verified
//
#include <hip/hip_runtime.h>
#include <math.h>

typedef __attribute__((ext_vector_type(2))) float v2f;
typedef __attribute__((ext_vector_type(8))) float v8f;

#define Bsz 256
#define Lsz 512
#define Dsz 256
#define Ksz 100

// ---------------------------------------------------------------------------
// emb_noise[b,n,:] = mean over 64 gathered rows
// ---------------------------------------------------------------------------
__global__ void noise_kernel(const int* __restrict__ noises,
                             const float* __restrict__ emb,
                             float* __restrict__ out) {
  __shared__ int s_idx[64];
  const int g = blockIdx.x;      // b*16 + n, 4096 blocks
  const int tid = threadIdx.x;   // 256 threads, one per dim d
  if (tid < 64) s_idx[tid] = noises[g * 64 + tid];
  __syncthreads();
  float acc = 0.f;
#pragma unroll 4
  for (int j = 0; j < 64; ++j)
    acc += emb[(size_t)s_idx[j] * Dsz + tid];
  out[(size_t)g * Dsz + tid] = acc * (1.0f / 64.0f);
}

// ---------------------------------------------------------------------------
// emb_sent[b,:] = mean over 512 gathered rows
// ---------------------------------------------------------------------------
__global__ void sent_kernel(const int* __restrict__ inputs,
                            const float* __restrict__ emb,
                            float* __restrict__ sent) {
  __shared__ int s_idx[Lsz];
  const int b = blockIdx.x;
  const int tid = threadIdx.x;   // 256 threads
  s_idx[tid]       = inputs[b * Lsz + tid];
  s_idx[tid + 256] = inputs[b * Lsz + tid + 256];
  __syncthreads();
  float acc = 0.f;
#pragma unroll 4
  for (int l = 0; l < Lsz; ++l)
    acc += emb[(size_t)s_idx[l] * Dsz + tid];
  sent[b * Dsz + tid] = acc * (1.0f / 512.0f);
}

// ---------------------------------------------------------------------------
// u[b,d] = sum_e emb_sent[b,e] * W_bil[d,e]        (M=256, N=256, K=256)
// f32 WMMA 16x16x4. A[m][k]=sent[m*256+k], B[k][n]=Wbil[n*256+k].
// Per ISA layout: lane<16 holds K={k0,k0+1}, lane>=16 holds K={k0+2,k0+3}.
// ---------------------------------------------------------------------------
__global__ void gemm_u_kernel(const float* __restrict__ sent,
                              const float* __restrict__ Wbil,
                              float* __restrict__ u) {
  const int wave = threadIdx.x >> 5;
  const int tile = blockIdx.x * (blockDim.x >> 5) + wave; // 0..255
  const int mt = tile >> 4, nt = tile & 15;
  const int lane = threadIdx.x & 31;
  const int half = lane >> 4;
  const int r16  = lane & 15;
  const int arow = mt * 16 + r16;   // A row (m)
  const int brow = nt * 16 + r16;   // B column (n) -> row of Wbil
  v8f c = {};
  for (int k0 = 0; k0 < 256; k0 += 4) {
    const int k = k0 + half * 2;
    v2f a = *(const v2f*)(sent + arow * 256 + k);
    v2f b = *(const v2f*)(Wbil + brow * 256 + k);
    c = __builtin_amdgcn_wmma_f32_16x16x4_f32(false, a, false, b,
                                              (short)0, c, false, false);
  }
  const int col = nt * 16 + r16;
#pragma unroll
  for (int r = 0; r < 8; ++r) {
    const int row = mt * 16 + r + half * 8;
    u[row * 256 + col] = c[r];
  }
}

// ---------------------------------------------------------------------------
// Fused attention: score -> softmax(L=512) -> weighted sum -> emb_attsent
// ---------------------------------------------------------------------------
__global__ void attn_kernel(const int* __restrict__ inputs,
                            const float* __restrict__ emb,
                            const float* __restrict__ u,
                            const float* __restrict__ bbil_p,
                            float* __restrict__ out_att) {
  __shared__ float s_u[Dsz];
  __shared__ float s_sc[Lsz];
  __shared__ float s_red[256];
  __shared__ int   s_idx[Lsz];
  const int b = blockIdx.x;
  const int tid = threadIdx.x;    // 256 threads = 8 waves
  s_u[tid]         = u[b * Dsz + tid];
  s_idx[tid]       = inputs[b * Lsz + tid];
  s_idx[tid + 256] = inputs[b * Lsz + tid + 256];
  __syncthreads();
  const float bbil = bbil_p[0];
  const int wave = tid >> 5, lane = tid & 31;

  // scores: one wave per row, lanes stride over D, wave32 xor-reduce
  for (int l = wave; l < Lsz; l += 8) {
    const float* row = emb + (size_t)s_idx[l] * Dsz;
    float acc = 0.f;
#pragma unroll
    for (int d = lane; d < Dsz; d += 32) acc += row[d] * s_u[d];
#pragma unroll
    for (int off = 16; off > 0; off >>= 1) acc += __shfl_xor(acc, off, 32);
    if (lane == 0) s_sc[l] = acc + bbil;
  }
  __syncthreads();

  // softmax over 512
  float m = fmaxf(s_sc[tid], s_sc[tid + 256]);
  s_red[tid] = m; __syncthreads();
  for (int s = 128; s > 0; s >>= 1) {
    if (tid < s) s_red[tid] = fmaxf(s_red[tid], s_red[tid + s]);
    __syncthreads();
  }
  const float mx = s_red[0]; __syncthreads();
  const float e0 = __expf(s_sc[tid] - mx);
  const float e1 = __expf(s_sc[tid + 256] - mx);
  s_sc[tid] = e0; s_sc[tid + 256] = e1;
  s_red[tid] = e0 + e1; __syncthreads();
  for (int s = 128; s > 0; s >>= 1) {
    if (tid < s) s_red[tid] += s_red[tid + s];
    __syncthreads();
  }
  const float inv = 1.0f / s_red[0];
  s_sc[tid] *= inv; s_sc[tid + 256] *= inv;
  __syncthreads();

  // weighted sum: thread d accumulates over all 512 rows (coalesced)
  float acc = 0.f;
#pragma unroll 4
  for (int l = 0; l < Lsz; ++l)
    acc += s_sc[l] * emb[(size_t)s_idx[l] * Dsz + tid];
  out_att[b * Dsz + tid] = acc;
}

// ---------------------------------------------------------------------------
// topic[b,:] = softmax(emb_attsent[b] @ W_enc^T + b_enc)   (K=100)
// ---------------------------------------------------------------------------
__global__ void topic_kernel(const float* __restrict__ att,
                             const float* __restrict__ Wenc,
                             const float* __restrict__ benc,
                             float* __restrict__ topic) {
  __shared__ float s_att[Dsz];
  __shared__ float s_red[128];
  const int b = blockIdx.x;
  const int tid = threadIdx.x;    // 128 threads
  s_att[tid]       = att[b * Dsz + tid];
  s_att[tid + 128] = att[b * Dsz + tid + 128];
  __syncthreads();
  float logit = -INFINITY;
  if (tid < Ksz) {
    const float* w = Wenc + tid * Dsz;
    float acc = benc[tid];
#pragma unroll 4
    for (int d = 0; d < Dsz; ++d) acc += w[d] * s_att[d];
    logit = acc;
  }
  s_red[tid] = logit; __syncthreads();
  for (int s = 64; s > 0; s >>= 1) {
    if (tid < s) s_red[tid] = fmaxf(s_red[tid], s_red[tid + s]);
    __syncthreads();
  }
  const float mx = s_red[0]; __syncthreads();
  const float e = (tid < Ksz) ? __expf(logit - mx) : 0.f;
  s_red[tid] = e; __syncthreads();
  for (int s = 64; s > 0; s >>= 1) {
    if (tid < s) s_red[tid] += s_red[tid + s];
    __syncthreads();
  }
  const float inv = 1.0f / s_red[0];
  if (tid < Ksz) topic[b * Ksz + tid] = e * inv;
}

// ---------------------------------------------------------------------------
// trans[b,d] = sum_k topic[b,k]*W_dec[d,k] + b_dec[d]   (M=256,N=256,K=100)
// f32 WMMA 16x16x4; K=100 = 25 exact k-steps of 4.
// ---------------------------------------------------------------------------
__global__ void gemm_dec_kernel(const float* __restrict__ topic,
                                const float* __restrict__ Wdec,
                                const float* __restrict__ bdec,
                                float* __restrict__ out) {
  const int wave = threadIdx.x >> 5;
  const int tile = blockIdx.x * (blockDim.x >> 5) + wave;
  const int mt = tile >> 4, nt = tile & 15;
  const int lane = threadIdx.x & 31;
  const int half = lane >> 4;
  const int r16  = lane & 15;
  const int arow = mt * 16 + r16;
  const int brow = nt * 16 + r16;
  v8f c = {};
  for (int k0 = 0; k0 < Ksz; k0 += 4) {
    const int k = k0 + half * 2;
    v2f a = *(const v2f*)(topic + arow * Ksz + k);
    v2f b = *(const v2f*)(Wdec  + brow * Ksz + k);
    c = __builtin_amdgcn_wmma_f32_16x16x4_f32(false, a, false, b,
                                              (short)0, c, false, false);
  }
  const int col = nt * 16 + r16;
  const float bias = bdec[col];
#pragma unroll
  for (int r = 0; r < 8; ++r) {
    const int row = mt * 16 + r + half * 8;
    out[row * 256 + col] = c[r] + bias;
  }
}

// ---------------------------------------------------------------------------
extern "C" void kernel_launch(void* const* d_in, const int* in_sizes, int n_in,
                              void* d_out, int out_size, void* d_ws, size_t ws_size,
                              hipStream_t stream) {
  const int*   inputs = (const int*)d_in[0];    // [256,512]
  const int*   noises = (const int*)d_in[1];    // [256,16,64]
  const float* emb    = (const float*)d_in[2];  // [50000,256]
  const float* Wbil   = (const float*)d_in[3];  // [256,256]
  const float* bbil   = (const float*)d_in[4];  // scalar
  const float* Wenc   = (const float*)d_in[5];  // [100,256]
  const float* benc   = (const float*)d_in[6];  // [100]
  const float* Wdec   = (const float*)d_in[7];  // [256,100]
  const float* bdec   = (const float*)d_in[8];  // [256]

  float* out       = (float*)d_out;
  float* out_att   = out;                       // [256,256]
  float* out_trans = out + 256 * 256;           // [256,256]
  float* out_noise = out + 2 * 256 * 256;       // [256,16,256]

  float* ws       = (float*)d_ws;
  float* ws_sent  = ws;                 // 65536 floats
  float* ws_u     = ws + 65536;         // 65536 floats
  float* ws_topic = ws + 131072;        // 25600 floats

  noise_kernel  <<<Bsz * 16, 256, 0, stream>>>(noises, emb, out_noise);
  sent_kernel   <<<Bsz,      256, 0, stream>>>(inputs, emb, ws_sent);
  gemm_u_kernel <<<32,       256, 0, stream>>>(ws_sent, Wbil, ws_u);
  attn_kernel   <<<Bsz,      256, 0, stream>>>(inputs, emb, ws_u, bbil, out_att);
  topic_kernel  <<<Bsz,      128, 0, stream>>>(out_att, Wenc, benc, ws_topic);
  gemm_dec_kernel<<<32,      256, 0, stream>>>(ws_topic, Wdec, bdec, out_trans);
}